// GraphSageE_2336462209765
// MI455X (gfx1250) — compile-verified
//
#include <hip/hip_runtime.h>

typedef __attribute__((ext_vector_type(2))) float v2f;
typedef __attribute__((ext_vector_type(8))) float v8f;

// ---------------------------------------------------------------------------
// Kernel 1: zero the accumulator (d_out, used as the sum buffer) and counts.
// ---------------------------------------------------------------------------
__global__ void gs_zero_kernel(float4* __restrict__ sums4, float* __restrict__ cnt,
                               int n4, int ncnt) {
    int i = blockIdx.x * blockDim.x + threadIdx.x;
    if (i < n4)   sums4[i] = make_float4(0.f, 0.f, 0.f, 0.f);
    if (i < ncnt) cnt[i]   = 0.f;
}

// ---------------------------------------------------------------------------
// Kernel 2: edge scatter. One wave32 per edge; lane L handles dims [2L,2L+1].
// Gather x[src] (two 128B lines, coalesced), global_atomic_add_f32 into
// sums[dst] (L2-resident, no-return atomics -> STOREcnt), count via lane 0.
// global_prefetch_b8 look-ahead on the next gathered row (random access).
// ---------------------------------------------------------------------------
__global__ void gs_scatter_kernel(const float* __restrict__ x,
                                  const int*   __restrict__ ei,
                                  float* __restrict__ sums,
                                  float* __restrict__ cnt,
                                  int nEdges) {
    const int t    = blockIdx.x * blockDim.x + threadIdx.x;
    const int e    = t >> 5;
    const int lane = t & 31;
    if (e >= nEdges) return;

    const int src = ei[e];            // row 0 of edge_index
    const int dst = ei[nEdges + e];   // row 1 of edge_index

    // Prefetch the source row of an edge ~64 waves ahead (2 lines per row).
    const int ep = e + 64;
    if (((lane & 15) == 0) && ep < nEdges) {
        const int srcp = ei[ep];
        __builtin_prefetch(x + (size_t)srcp * 64 + (size_t)(lane >> 4) * 32, 0, 0);
    }

    const float2 v = *(const float2*)(x + (size_t)src * 64 + lane * 2);
    float* o = sums + (size_t)dst * 64 + lane * 2;
    atomicAdd(o,     v.x);
    atomicAdd(o + 1, v.y);
    if (lane == 0) atomicAdd(cnt + dst, 1.0f);
}

// ---------------------------------------------------------------------------
// Kernel 3: mean + L2-normalize + 2x + ReLU, in place, one wave32 per 16 nodes.
// Squared row norms come from diag(M * M^T) computed with 16 accumulating
// V_WMMA_F32_16X16X4_F32 ops. The f32 A-layout (16x4) and B-layout (4x16) are
// transposes of each other, so the same v2f register pair feeds both A and B.
// ---------------------------------------------------------------------------
__global__ void gs_normalize_kernel(float* __restrict__ io,
                                    const float* __restrict__ cnt,
                                    int nNodes) {
    const int t    = blockIdx.x * blockDim.x + threadIdx.x;
    const int wave = t >> 5;
    const int lane = t & 31;
    const int base = wave * 16;
    if (base >= nNodes) return;          // wave-uniform: EXEC stays all-1s for WMMA

    const int row  = lane & 15;          // node within the 16-block (M index)
    const int hi   = lane >> 4;          // K half: lanes 0-15 -> K=0,1 ; 16-31 -> K=2,3
    const int node = base + row;

    const float c   = cnt[node];
    const float inv = 1.0f / fmaxf(c, 1.0f);

    float* rowp = io + (size_t)node * 64 + hi * 2;

    v2f m[16];
    v8f acc = {};
#pragma unroll
    for (int k = 0; k < 16; ++k) {
        v2f v = *(const v2f*)(rowp + k * 4);   // cols 4k + 2*hi .. +1
        v.x *= inv;
        v.y *= inv;
        m[k] = v;
        // acc += A_chunk * A_chunk^T   (A = B register-wise for f32 16x16x4)
        acc = __builtin_amdgcn_wmma_f32_16x16x4_f32(
            /*neg_a=*/false, v, /*neg_b=*/false, v,
            /*c_mod=*/(short)0, acc, /*reuse_a=*/false, /*reuse_b=*/false);
    }

    // Extract diag: G[v][v] lives in acc[v] lane v; G[v+8][v+8] in acc[v] lane v+24.
    float nsq = 0.f;
#pragma unroll
    for (int v = 0; v < 8; ++v) {
        const float d0 = __shfl(acc[v], v, 32);
        const float d1 = __shfl(acc[v], v + 24, 32);
        nsq = (row == v)     ? d0 : nsq;
        nsq = (row == v + 8) ? d1 : nsq;
    }

    const float s = 2.0f / fmaxf(sqrtf(nsq), 1e-12f);  // relu(2 * mean / max(||mean||,eps))
#pragma unroll
    for (int k = 0; k < 16; ++k) {
        v2f v = m[k];
        v2f o;
        o.x = fmaxf(v.x * s, 0.0f);
        o.y = fmaxf(v.y * s, 0.0f);
        *(v2f*)(rowp + k * 4) = o;
    }
}

// ---------------------------------------------------------------------------
extern "C" void kernel_launch(void* const* d_in, const int* in_sizes, int n_in,
                              void* d_out, int out_size, void* d_ws, size_t ws_size,
                              hipStream_t stream) {
    const float* x  = (const float*)d_in[0];
    const int*   ei = (const int*)d_in[1];
    // d_in[2] = edge_weights: unused by the reference computation.

    float* out = (float*)d_out;      // doubles as the scatter-sum accumulator
    float* cnt = (float*)d_ws;       // 100000 floats = 400 KB of workspace

    const int dim    = 64;
    const int nNodes = in_sizes[0] / dim;     // 100000
    const int nEdges = in_sizes[1] / 2;       // 1250000

    // 1) zero accumulator + counts
    {
        const int n4 = nNodes * (dim / 4);    // float4 count for sums
        const int nb = (n4 + 255) / 256;
        gs_zero_kernel<<<nb, 256, 0, stream>>>((float4*)out, cnt, n4, nNodes);
    }

    // 2) edge scatter: one wave32 per edge
    {
        const long long threads = (long long)nEdges * 32;
        const int nb = (int)((threads + 255) / 256);
        gs_scatter_kernel<<<nb, 256, 0, stream>>>(x, ei, out, cnt, nEdges);
    }

    // 3) mean + normalize + relu (WMMA Gram-diagonal), one wave32 per 16 nodes
    {
        const int nWaves = (nNodes + 15) / 16;
        const long long threads = (long long)nWaves * 32;
        const int nb = (int)((threads + 255) / 256);
        gs_normalize_kernel<<<nb, 256, 0, stream>>>(out, cnt, nNodes);
    }
}